// my_proposal_layer_83648783056897
// MI455X (gfx1250) — compile-verified
//
#include <hip/hip_runtime.h>
#include <hip/hip_bf16.h>
#include <math.h>

// ---------------- problem constants (match reference) ----------------
#define Bq   32
#define Aq   9
#define Hq   128
#define Wq   128
#define Nq   (Aq * Hq * Wq)       // 147456 anchors per image
#define PRE  6000                 // PRE_NMS_TOP_N
#define POST 300                  // POST_NMS_TOP_N
#define SORTN 8192                // next pow2 >= PRE for bitonic sort
#define NBIN 65536
#define ALIVE_WORDS ((PRE + 31) / 32)   // 188

// Base anchors for generate_anchors(16,[0.5,1,2],[8,16,32]) (precomputed,
// identical to the numpy reference including np.round half-to-even).
__constant__ float c_base[9][4] = {
    {-84.f,  -40.f,   99.f,  55.f},
    {-176.f, -88.f,  191.f, 103.f},
    {-360.f, -184.f, 375.f, 199.f},
    {-56.f,  -56.f,   71.f,  71.f},
    {-120.f, -120.f, 135.f, 135.f},
    {-248.f, -248.f, 263.f, 263.f},
    {-36.f,  -80.f,   51.f,  95.f},
    {-80.f,  -168.f,  95.f, 183.f},
    {-168.f, -344.f, 183.f, 359.f}};

// ---------------- CDNA5 async global->LDS path (guarded) ----------------
#if defined(__HIP_DEVICE_COMPILE__)
  #if __has_builtin(__builtin_amdgcn_global_load_async_to_lds_b128)
    #define HAVE_ASYNC_LDS 1
    typedef int v4i_async __attribute__((vector_size(16)));
    typedef __attribute__((address_space(1))) v4i_async *g_v4i_ptr;
    typedef __attribute__((address_space(3))) v4i_async *l_v4i_ptr;
    #define ASYNC_COPY16(gp, lp)                                              \
      __builtin_amdgcn_global_load_async_to_lds_b128((g_v4i_ptr)(gp),         \
                                                     (l_v4i_ptr)(lp), 0, 0)
  #endif
  #if __has_builtin(__builtin_amdgcn_s_wait_asynccnt)
    #define WAIT_ASYNC() __builtin_amdgcn_s_wait_asynccnt(0)
  #elif defined(HAVE_ASYNC_LDS)
    #define WAIT_ASYNC() asm volatile("s_wait_asynccnt 0" ::: "memory")
  #endif
#endif
#ifndef ASYNC_COPY16
  #define ASYNC_COPY16(gp, lp) (*(float4 *)(lp) = *(const float4 *)(gp))
#endif
#ifndef WAIT_ASYNC
  #define WAIT_ASYNC() do {} while (0)
#endif

// ---------------- helpers ----------------
// Monotonic float->uint key: larger float => larger unsigned key.
__device__ __forceinline__ unsigned key_of(float s) {
  unsigned u = __float_as_uint(s);
  return (u & 0x80000000u) ? ~u : (u | 0x80000000u);
}

// Decode + clip one proposal (identical math/order to the reference).
__device__ __forceinline__ void decode_box(const float *__restrict__ bbx,
                                           const float *__restrict__ info,
                                           int b, int i, float box[4]) {
  int a = i % Aq;
  int pix = i / Aq;
  int wp = pix % Wq, hp = pix / Wq;
  float shx = (float)(wp * 16), shy = (float)(hp * 16);
  float ax1 = c_base[a][0] + shx, ay1 = c_base[a][1] + shy;
  float ax2 = c_base[a][2] + shx, ay2 = c_base[a][3] + shy;
  float aw = ax2 - ax1 + 1.0f, ah = ay2 - ay1 + 1.0f;
  float cx = ax1 + 0.5f * aw, cy = ay1 + 0.5f * ah;
  size_t dbase = ((size_t)b * (4 * Aq) + 4 * a) * (size_t)(Hq * Wq) +
                 (size_t)hp * Wq + wp;
  float dx = bbx[dbase];
  float dy = bbx[dbase + (size_t)(Hq * Wq)];
  float dw = bbx[dbase + (size_t)(2 * Hq * Wq)];
  float dh = bbx[dbase + (size_t)(3 * Hq * Wq)];
  float pcx = dx * aw + cx, pcy = dy * ah + cy;
  float pw = expf(dw) * aw, ph = expf(dh) * ah;
  float imh = info[b * 3 + 0], imw = info[b * 3 + 1];
  box[0] = fminf(fmaxf(pcx - 0.5f * pw, 0.0f), imw - 1.0f);
  box[1] = fminf(fmaxf(pcy - 0.5f * ph, 0.0f), imh - 1.0f);
  box[2] = fminf(fmaxf(pcx + 0.5f * pw, 0.0f), imw - 1.0f);
  box[3] = fminf(fmaxf(pcy + 0.5f * ph, 0.0f), imh - 1.0f);
}

// ---------------- kernels ----------------
__global__ void zero_ws_k(unsigned *__restrict__ p, long long n) {
  long long t = (long long)blockIdx.x * blockDim.x + threadIdx.x;
  if (t < n) p[t] = 0u;
}

// Fused decode + min-size filter + sortable key + coarse (top-16-bit) histogram.
__global__ void decode_filter_k(const float *__restrict__ cls,
                                const float *__restrict__ bbx,
                                const float *__restrict__ info,
                                unsigned *__restrict__ keys,
                                unsigned *__restrict__ hist1) {
  long long t = (long long)blockIdx.x * blockDim.x + threadIdx.x;
  if (t >= (long long)Bq * Nq) return;
  int b = (int)(t / Nq), i = (int)(t % Nq);
  float box[4];
  decode_box(bbx, info, b, i, box);
  float wsz = box[2] - box[0] + 1.0f, hsz = box[3] - box[1] + 1.0f;
  float minsz = 16.0f * info[b * 3 + 2];
  int a = i % Aq;
  int pix = i / Aq;
  int wp = pix % Wq, hp = pix / Wq;
  float sc = cls[(((size_t)b * (2 * Aq) + Aq + a) * Hq + hp) * Wq + wp];
  if (!(wsz >= minsz && hsz >= minsz)) sc = -1e30f;
  unsigned key = key_of(sc);
  keys[(size_t)b * Nq + i] = key;
  atomicAdd(&hist1[(size_t)b * NBIN + (key >> 16)], 1u);
}

// Per-image: find coarse cutoff bin t so count(top16 > t) < PRE <= count(top16 >= t)
__global__ __launch_bounds__(1024) void coarse_cut_k(const unsigned *__restrict__ hist,
                                                     unsigned *__restrict__ meta) {
  __shared__ unsigned part[1024];
  int b = blockIdx.x;
  const unsigned *h = hist + (size_t)b * NBIN;
  unsigned s = 0;
  int base = threadIdx.x * 64;
  for (int c = 0; c < 64; ++c) s += h[base + c];
  part[threadIdx.x] = s;
  __syncthreads();
  if (threadIdx.x == 0) {
    unsigned acc = 0;
    int chunk = 0;
    for (int c = 1023; c >= 0; --c) {
      if (acc + part[c] >= (unsigned)PRE) { chunk = c; break; }
      acc += part[c];
    }
    unsigned tbin = (unsigned)(chunk * 64), S = acc;
    for (int bin = chunk * 64 + 63; bin >= chunk * 64; --bin) {
      unsigned v = h[bin];
      if (S + v >= (unsigned)PRE) { tbin = (unsigned)bin; break; }
      S += v;
    }
    meta[b * 16 + 0] = tbin;   // coarse bin
    meta[b * 16 + 1] = S;      // count strictly above bin
  }
}

// Histogram of low 16 key bits, restricted to the coarse cutoff bin.
__global__ void fine_hist_k(const unsigned *__restrict__ keys,
                            const unsigned *__restrict__ meta,
                            unsigned *__restrict__ hist2) {
  long long t = (long long)blockIdx.x * blockDim.x + threadIdx.x;
  if (t >= (long long)Bq * Nq) return;
  int b = (int)(t / Nq);
  unsigned key = keys[t];
  if ((key >> 16) == meta[b * 16 + 0])
    atomicAdd(&hist2[(size_t)b * NBIN + (key & 0xFFFFu)], 1u);
}

// Per-image: exact cutoff key and count of strictly-greater keys.
__global__ __launch_bounds__(1024) void fine_cut_k(const unsigned *__restrict__ hist2,
                                                   unsigned *__restrict__ meta) {
  __shared__ unsigned part[1024];
  int b = blockIdx.x;
  const unsigned *h = hist2 + (size_t)b * NBIN;
  unsigned s = 0;
  int base = threadIdx.x * 64;
  for (int c = 0; c < 64; ++c) s += h[base + c];
  part[threadIdx.x] = s;
  __syncthreads();
  if (threadIdx.x == 0) {
    unsigned Sab = meta[b * 16 + 1];
    unsigned need = (unsigned)PRE - Sab;   // >= 1 by construction
    unsigned acc = 0;
    int chunk = 0;
    for (int c = 1023; c >= 0; --c) {
      if (acc + part[c] >= need) { chunk = c; break; }
      acc += part[c];
    }
    unsigned u = (unsigned)(chunk * 64), F = acc;
    for (int bin = chunk * 64 + 63; bin >= chunk * 64; --bin) {
      unsigned v = h[bin];
      if (F + v >= need) { u = (unsigned)bin; break; }
      F += v;
    }
    unsigned t16 = meta[b * 16 + 0];
    meta[b * 16 + 2] = (t16 << 16) | u;  // cutoff key
    meta[b * 16 + 3] = Sab + F;          // nGreater = count(key > cutoff)
  }
}

// Gather exactly PRE (key, idx) pairs per image (unsorted).
__global__ void gather_k(const unsigned *__restrict__ keys,
                         unsigned *__restrict__ meta,
                         unsigned *__restrict__ ck, unsigned *__restrict__ ci) {
  long long t = (long long)blockIdx.x * blockDim.x + threadIdx.x;
  if (t >= (long long)Bq * Nq) return;
  int b = (int)(t / Nq), i = (int)(t % Nq);
  unsigned key = keys[t];
  unsigned cut = meta[b * 16 + 2];
  if (key > cut) {
    unsigned p = atomicAdd(&meta[b * 16 + 4], 1u);
    ck[(size_t)b * PRE + p] = key;
    ci[(size_t)b * PRE + p] = (unsigned)i;
  } else if (key == cut) {
    unsigned nG = meta[b * 16 + 3];
    unsigned e = atomicAdd(&meta[b * 16 + 5], 1u);
    if (nG + e < (unsigned)PRE) {
      ck[(size_t)b * PRE + nG + e] = key;
      ci[(size_t)b * PRE + nG + e] = (unsigned)i;
    }
  }
}

__device__ __forceinline__ bool beforeKI(unsigned ka, unsigned ia, unsigned kb,
                                         unsigned ib) {
  // descending key, ascending index tie-break (== lax.top_k tie-break)
  return (ka > kb) || (ka == kb && ia < ib);
}

// In-LDS bitonic sort of PRE pairs (padded to SORTN). 64 KB static LDS.
__global__ __launch_bounds__(1024) void sort_cand_k(unsigned *__restrict__ ck,
                                                    unsigned *__restrict__ ci) {
  __shared__ unsigned sk[SORTN];
  __shared__ unsigned si[SORTN];
  int b = blockIdx.x;
  for (int i = threadIdx.x; i < SORTN; i += 1024) {
    if (i < PRE) {
      sk[i] = ck[(size_t)b * PRE + i];
      si[i] = ci[(size_t)b * PRE + i];
    } else {
      sk[i] = 0u;
      si[i] = 0x7FFFFFFFu;
    }
  }
  __syncthreads();
  for (unsigned k = 2; k <= SORTN; k <<= 1) {
    for (unsigned j = k >> 1; j > 0; j >>= 1) {
      for (unsigned i = threadIdx.x; i < SORTN; i += 1024) {
        unsigned ixj = i ^ j;
        if (ixj > i) {
          unsigned ka = sk[i], ia = si[i], kb = sk[ixj], ib = si[ixj];
          bool descBlk = ((i & k) == 0);
          bool doSwap = descBlk ? beforeKI(kb, ib, ka, ia)
                                : beforeKI(ka, ia, kb, ib);
          if (doSwap) {
            sk[i] = kb; si[i] = ib;
            sk[ixj] = ka; si[ixj] = ia;
          }
        }
      }
      __syncthreads();
    }
  }
  for (int i = threadIdx.x; i < PRE; i += 1024) {
    ck[(size_t)b * PRE + i] = sk[i];
    ci[(size_t)b * PRE + i] = si[i];
  }
}

// Re-decode the 6000 selected proposals per image.
__global__ void decode_cand_k(const float *__restrict__ bbx,
                              const float *__restrict__ info,
                              const unsigned *__restrict__ ci,
                              float *__restrict__ cbox) {
  int t = blockIdx.x * blockDim.x + threadIdx.x;
  if (t >= Bq * PRE) return;
  int b = t / PRE;
  unsigned idx = ci[t];
  float box[4] = {0.f, 0.f, 0.f, 0.f};
  if (idx < (unsigned)Nq) decode_box(bbx, info, b, (int)idx, box);
  ((float4 *)cbox)[t] = make_float4(box[0], box[1], box[2], box[3]);
}

// Greedy NMS over sorted candidates (== sequential argmax NMS of the
// reference, since scores are descending and suppression writes NEG_INF).
// All 6000 boxes live in LDS (96 KB — CDNA5 320KB/WGP), staged via the
// gfx1250 async global->LDS path.
__global__ __launch_bounds__(1024) void nms_out_k(const float *__restrict__ cbox,
                                                  const unsigned *__restrict__ ck,
                                                  float *__restrict__ out) {
  __shared__ float sb[PRE * 4];                 // 96000 B
  __shared__ unsigned alive[ALIVE_WORDS];       // 188 words
  __shared__ unsigned pickS;
  __shared__ unsigned cursorS;
  int b = blockIdx.x;
  const float *src = cbox + (size_t)b * PRE * 4;

  for (int r = threadIdx.x; r < PRE; r += 1024)
    ASYNC_COPY16(src + (size_t)r * 4, &sb[r * 4]);
  WAIT_ASYNC();

  for (int w = threadIdx.x; w < ALIVE_WORDS; w += 1024) alive[w] = 0u;
  if (threadIdx.x == 0) cursorS = 0u;
  __syncthreads();

  const unsigned NEGKEY = key_of(-1e30f);
  for (int r = threadIdx.x; r < PRE; r += 1024)
    if (ck[(size_t)b * PRE + r] > NEGKEY)
      atomicOr(&alive[r >> 5], 1u << (r & 31));

  // pre-fill output: id column = b, boxes zero (matches invalid rows).
  for (int k = threadIdx.x; k < POST; k += 1024) {
    float *o = out + ((size_t)b * POST + k) * 5;
    o[0] = (float)b; o[1] = 0.f; o[2] = 0.f; o[3] = 0.f; o[4] = 0.f;
  }
  __syncthreads();

  for (int k = 0; k < POST; ++k) {
    if (threadIdx.x == 0) {
      unsigned r = 0xFFFFFFFFu;
      unsigned cur = cursorS;
      unsigned mask = ~0u << (cur & 31);
      for (int w = (int)(cur >> 5); w < ALIVE_WORDS; ++w) {
        unsigned bits = alive[w] & mask;
        mask = ~0u;
        if (bits) { r = (unsigned)(w * 32 + (__ffs(bits) - 1)); break; }
      }
      pickS = r;
      if (r != 0xFFFFFFFFu) cursorS = r + 1u;
    }
    __syncthreads();
    unsigned r = pickS;
    if (r == 0xFFFFFFFFu) break;  // uniform across block
    float px1 = sb[r * 4 + 0], py1 = sb[r * 4 + 1];
    float px2 = sb[r * 4 + 2], py2 = sb[r * 4 + 3];
    if (threadIdx.x == 0) {
      float *o = out + ((size_t)b * POST + k) * 5;
      o[1] = px1; o[2] = py1; o[3] = px2; o[4] = py2;
    }
    float parea = (px2 - px1 + 1.0f) * (py2 - py1 + 1.0f);
    for (int j = threadIdx.x; j < PRE; j += 1024) {
      if (alive[j >> 5] & (1u << (j & 31))) {
        float x1 = sb[j * 4 + 0], y1 = sb[j * 4 + 1];
        float x2 = sb[j * 4 + 2], y2 = sb[j * 4 + 3];
        float xx1 = fmaxf(px1, x1), yy1 = fmaxf(py1, y1);
        float xx2 = fminf(px2, x2), yy2 = fminf(py2, y2);
        float iw = fmaxf(xx2 - xx1 + 1.0f, 0.0f);
        float ih = fmaxf(yy2 - yy1 + 1.0f, 0.0f);
        float inter = iw * ih;
        float area = (x2 - x1 + 1.0f) * (y2 - y1 + 1.0f);
        float iou = inter / fmaxf(parea + area - inter, 1e-6f);
        if (iou > 0.7f) atomicAnd(&alive[j >> 5], ~(1u << (j & 31)));
      }
    }
    __syncthreads();
  }
}

// ---------------- launcher ----------------
extern "C" void kernel_launch(void *const *d_in, const int *in_sizes, int n_in,
                              void *d_out, int out_size, void *d_ws,
                              size_t ws_size, hipStream_t stream) {
  (void)in_sizes; (void)n_in; (void)out_size; (void)ws_size;
  const float *cls  = (const float *)d_in[0];  // (32, 18, 128, 128)
  const float *bbx  = (const float *)d_in[1];  // (32, 36, 128, 128)
  const float *info = (const float *)d_in[2];  // (32, 3)
  float *out = (float *)d_out;                 // (32, 300, 5)

  char *w = (char *)d_ws;
  size_t off = 0;
  auto alloc = [&](size_t bytes) -> void * {
    void *p = (void *)(w + off);
    off += (bytes + 255) & ~(size_t)255;
    return p;
  };
  unsigned *keys  = (unsigned *)alloc((size_t)Bq * Nq * 4);       // 18.9 MB
  unsigned *hist1 = (unsigned *)alloc((size_t)Bq * NBIN * 4);     // 8 MB
  unsigned *hist2 = (unsigned *)alloc((size_t)Bq * NBIN * 4);     // 8 MB
  unsigned *meta  = (unsigned *)alloc((size_t)Bq * 16 * 4);       // 2 KB
  unsigned *ck    = (unsigned *)alloc((size_t)Bq * PRE * 4);
  unsigned *ci    = (unsigned *)alloc((size_t)Bq * PRE * 4);
  float    *cbox  = (float *)alloc((size_t)Bq * PRE * 16);        // 3 MB

  // hist1, hist2, meta are contiguous: zero them in one pass (every call —
  // they are accumulated into, and the harness replays without re-poisoning).
  long long nz = (long long)Bq * NBIN * 2 + Bq * 16;
  zero_ws_k<<<(unsigned)((nz + 255) / 256), 256, 0, stream>>>(hist1, nz);

  long long totBN = (long long)Bq * Nq;
  unsigned gBN = (unsigned)((totBN + 255) / 256);

  decode_filter_k<<<gBN, 256, 0, stream>>>(cls, bbx, info, keys, hist1);
  coarse_cut_k<<<Bq, 1024, 0, stream>>>(hist1, meta);
  fine_hist_k<<<gBN, 256, 0, stream>>>(keys, meta, hist2);
  fine_cut_k<<<Bq, 1024, 0, stream>>>(hist2, meta);
  gather_k<<<gBN, 256, 0, stream>>>(keys, meta, ck, ci);
  sort_cand_k<<<Bq, 1024, 0, stream>>>(ck, ci);
  decode_cand_k<<<(Bq * PRE + 255) / 256, 256, 0, stream>>>(bbx, info, ci, cbox);
  nms_out_k<<<Bq, 1024, 0, stream>>>(cbox, ck, out);
}